// GCN_4_layer_edge_weight_fc_45311904973178
// MI455X (gfx1250) — compile-verified
//
#include <hip/hip_runtime.h>

typedef float v2f __attribute__((ext_vector_type(2)));
typedef float v8f __attribute__((ext_vector_type(8)));
typedef unsigned int u32;
typedef u32 v4u __attribute__((ext_vector_type(4)));
typedef u32 v8u __attribute__((ext_vector_type(8)));

#define D_FEAT 64

__global__ __launch_bounds__(256) void zero_kernel(float* __restrict__ p, long n) {
    long i = (long)blockIdx.x * blockDim.x + threadIdx.x;
    if (i < n) p[i] = 0.0f;
}

__global__ __launch_bounds__(256) void degree_kernel(const int* __restrict__ src,
                                                     const int* __restrict__ dst,
                                                     float* deg_out, float* deg_in, int E) {
    int e = blockIdx.x * blockDim.x + threadIdx.x;
    if (e >= E) return;
    __hip_atomic_fetch_add(&deg_out[src[e]], 1.0f, __ATOMIC_RELAXED, __HIP_MEMORY_SCOPE_AGENT);
    __hip_atomic_fetch_add(&deg_in[dst[e]],  1.0f, __ATOMIC_RELAXED, __HIP_MEMORY_SCOPE_AGENT);
}

__global__ __launch_bounds__(256) void norm_kernel(const float* __restrict__ deg_out,
                                                   const float* __restrict__ deg_in,
                                                   float* __restrict__ norm_s,
                                                   float* __restrict__ norm_d, int N) {
    int i = blockIdx.x * blockDim.x + threadIdx.x;
    if (i >= N) return;
    norm_s[i] = rsqrtf(fmaxf(deg_out[i], 1.0f));
    norm_d[i] = rsqrtf(fmaxf(deg_in[i],  1.0f));
}

// One thread per (edge, 4-float chunk): 16 threads cover one edge's 64 features.
// m = h[src]*norm_s[src]*ew scatter-added into agg[dst] (L2-resident on MI455X).
__global__ __launch_bounds__(256) void scatter_kernel(const float* __restrict__ h,
                                                      const int* __restrict__ src,
                                                      const int* __restrict__ dst,
                                                      const float* __restrict__ ew,
                                                      const float* __restrict__ norm_s,
                                                      float* __restrict__ agg, int E) {
    long gid = (long)blockIdx.x * blockDim.x + threadIdx.x;
    int e = (int)(gid >> 4);
    if (e >= E) return;
    int c = ((int)gid & 15) << 2;
    int s = src[e];
    int d = dst[e];
    float w = ew[e] * norm_s[s];
    const float4 v = *(const float4*)(h + (long)s * D_FEAT + c);
    float* o = agg + (long)d * D_FEAT + c;
    __hip_atomic_fetch_add(o + 0, v.x * w, __ATOMIC_RELAXED, __HIP_MEMORY_SCOPE_AGENT);
    __hip_atomic_fetch_add(o + 1, v.y * w, __ATOMIC_RELAXED, __HIP_MEMORY_SCOPE_AGENT);
    __hip_atomic_fetch_add(o + 2, v.z * w, __ATOMIC_RELAXED, __HIP_MEMORY_SCOPE_AGENT);
    __hip_atomic_fetch_add(o + 3, v.w * w, __ATOMIC_RELAXED, __HIP_MEMORY_SCOPE_AGENT);
}

// Stage nElems f32 from global into LDS via the Tensor Data Mover.
// D# built in SGPRs: group0 = {count=1 | lds_addr | global_addr | type=2},
// group1 = 1-D tensor: data_size=4B, tensor_dim0=tile_dim0=nElems, stride=nElems.
// Issue once (one wave), wait TENSORcnt==0, caller barriers.
__device__ __forceinline__ void tdm_stage_to_lds(u32 lds_off, const float* gsrc, int nElems) {
    unsigned long long ga = (unsigned long long)(uintptr_t)gsrc;
    v4u g0;
    g0.x = 1u;                                        // count=1, user descriptor
    g0.y = lds_off;                                   // lds_addr (bytes)
    g0.z = (u32)ga;                                   // global_addr[31:0]
    g0.w = ((u32)(ga >> 32) & 0x01FFFFFFu) | (2u << 30);  // addr[56:32] | type=2
    v8u g1;
    g1[0] = 2u << 16;                                 // data_size=4B, no multicast
    g1[1] = ((u32)nElems & 0xFFFFu) << 16;            // tensor_dim0[15:0]  (bits 63:48)
    g1[2] = ((u32)nElems >> 16) | (1u << 16);         // tensor_dim0[31:16] | tensor_dim1=1
    g1[3] = ((u32)nElems & 0xFFFFu) << 16;            // tile_dim0 (bits 127:112)
    g1[4] = 0u;                                       // tile_dim1=0, tile_dim2=0
    g1[5] = (u32)nElems;                              // tensor_dim0_stride[31:0]
    g1[6] = 0u;
    g1[7] = 0u;
    asm volatile("tensor_load_to_lds %0, %1" :: "s"(g0), "s"(g1) : "memory");
    __builtin_amdgcn_s_wait_tensorcnt(0);
}

// out[n, OUTC] = relu?( (rowScale.*A) @ W + bias + resid )
// A: [nRows,64] f32, W: [64,OUTC] staged in LDS via TDM; K=64 via
// V_WMMA_F32_16X16X4_F32. One wave per 16x16 output tile; 8 waves per block.
template <int OUTC>
__global__ __launch_bounds__(256) void wmma_gemm_kernel(const float* __restrict__ A,
                                                        const float* __restrict__ W,
                                                        const float* __restrict__ bias,
                                                        const float* __restrict__ rowScale,
                                                        const float* __restrict__ resid,
                                                        float* __restrict__ out,
                                                        int nRows, int doRelu) {
    __shared__ float sW[D_FEAT * OUTC];

    // Block-wide staging of W into LDS (one TDM issue from wave 0).
    if (threadIdx.x < 32) {
        tdm_stage_to_lds((u32)(uintptr_t)&sW[0], W, D_FEAT * OUTC);
    }
    __syncthreads();

    const int lane = threadIdx.x & 31;
    const int wave = threadIdx.x >> 5;
    const int half = lane >> 4;          // 0: lanes 0-15, 1: lanes 16-31
    const int lm   = lane & 15;
    const int nct  = OUTC >> 4;          // column tiles of 16
    const int gwid = blockIdx.x * 8 + wave;
    const int tiles = (nRows >> 4) * nct;   // nRows divisible by 16 (N=100000)
    if (gwid >= tiles) return;              // whole-wave exit: EXEC all-1s for WMMA

    const int rowTile = gwid / nct;
    const int colTile = gwid - rowTile * nct;
    const int m0 = rowTile << 4;
    const int n0 = colTile << 4;

    // A fragment: row = lane%16 fixed; K = k + i + 2*half
    const int rowM = m0 + lm;
    const float scale = rowScale ? rowScale[rowM] : 1.0f;
    const float* Arow = A + (long)rowM * D_FEAT + 2 * half;
    // B fragment from LDS: col = n0 + lane%16 fixed; K = k + i + 2*half
    const float* sWcol = &sW[(2 * half) * OUTC + n0 + lm];

    v8f c = {0.f, 0.f, 0.f, 0.f, 0.f, 0.f, 0.f, 0.f};
#pragma unroll
    for (int k = 0; k < D_FEAT; k += 4) {
        float2 av = *(const float2*)(Arow + k);
        v2f a, b;
        a.x = av.x * scale;
        a.y = av.y * scale;
        b.x = sWcol[k * OUTC];           // immediate ds offsets (OUTC constexpr)
        b.y = sWcol[(k + 1) * OUTC];
        c = __builtin_amdgcn_wmma_f32_16x16x4_f32(false, a, false, b, (short)0, c,
                                                  false, false);
    }

    const int col = n0 + lm;
    const float bv = bias[col];
#pragma unroll
    for (int i = 0; i < 8; i++) {
        int row = m0 + i + 8 * half;     // C/D layout: VGPR i -> M = i + 8*half
        float v = c[i] + bv;
        if (resid) v += resid[(long)row * OUTC + col];
        if (doRelu) v = fmaxf(v, 0.0f);
        out[(long)row * OUTC + col] = v;
    }
}

extern "C" void kernel_launch(void* const* d_in, const int* in_sizes, int n_in,
                              void* d_out, int out_size, void* d_ws, size_t ws_size,
                              hipStream_t stream) {
    const float* x   = (const float*)d_in[0];
    const int*   src = (const int*)d_in[1];
    const int*   dst = (const int*)d_in[2];
    const float* ew  = (const float*)d_in[3];
    const float* W1 = (const float*)d_in[4];   const float* b1 = (const float*)d_in[5];
    const float* W2 = (const float*)d_in[6];   const float* b2 = (const float*)d_in[7];
    const float* W3 = (const float*)d_in[8];   const float* b3 = (const float*)d_in[9];
    const float* W4 = (const float*)d_in[10];  const float* b4 = (const float*)d_in[11];
    const float* Wr = (const float*)d_in[12];  const float* br = (const float*)d_in[13];
    const float* Wo = (const float*)d_in[14];  const float* bo = (const float*)d_in[15];

    const int N = in_sizes[0] / D_FEAT;   // 100000
    const int E = in_sizes[1];            // 1600000
    const int C = in_sizes[15];           // 32

    float* ws = (float*)d_ws;
    float* deg_out = ws;                       // [N]
    float* deg_in  = ws + (long)N;             // [N]
    float* norm_s  = ws + 2L * N;              // [N]
    float* norm_d  = ws + 3L * N;              // [N]
    float* res = ws + 4L * N;                  // [N,64]
    float* agg = res + (long)N * D_FEAT;       // [N,64]
    float* h0  = agg + (long)N * D_FEAT;       // [N,64]
    float* h1  = h0  + (long)N * D_FEAT;       // [N,64]

    const int T = 256;
    const int degBlocks  = (E + T - 1) / T;
    const int nodeBlocks = (N + T - 1) / T;
    const long aggElems  = (long)N * D_FEAT;
    const int zeroAggBlocks = (int)((aggElems + T - 1) / T);
    const long scatThreads = (long)E * 16;
    const int scatBlocks = (int)((scatThreads + T - 1) / T);
    const int gemm64Blocks = ((N >> 4) * (D_FEAT >> 4) + 7) / 8;  // 8 waves/block
    const int gemmOutBlocks = ((N >> 4) * (C >> 4) + 7) / 8;

    // Degrees + symmetric norms
    zero_kernel<<<(2 * N + T - 1) / T, T, 0, stream>>>(deg_out, 2L * N);
    degree_kernel<<<degBlocks, T, 0, stream>>>(src, dst, deg_out, deg_in, E);
    norm_kernel<<<nodeBlocks, T, 0, stream>>>(deg_out, deg_in, norm_s, norm_d, N);

    // Residual: res = x @ Wr + br
    wmma_gemm_kernel<D_FEAT><<<gemm64Blocks, T, 0, stream>>>(x, Wr, br, nullptr,
                                                             nullptr, res, N, 0);

    // Layer 1: x -> h0
    zero_kernel<<<zeroAggBlocks, T, 0, stream>>>(agg, aggElems);
    scatter_kernel<<<scatBlocks, T, 0, stream>>>(x, src, dst, ew, norm_s, agg, E);
    wmma_gemm_kernel<D_FEAT><<<gemm64Blocks, T, 0, stream>>>(agg, W1, b1, norm_d,
                                                             nullptr, h0, N, 1);
    // Layer 2: h0 -> h1
    zero_kernel<<<zeroAggBlocks, T, 0, stream>>>(agg, aggElems);
    scatter_kernel<<<scatBlocks, T, 0, stream>>>(h0, src, dst, ew, norm_s, agg, E);
    wmma_gemm_kernel<D_FEAT><<<gemm64Blocks, T, 0, stream>>>(agg, W2, b2, norm_d,
                                                             nullptr, h1, N, 1);
    // Layer 3: h1 -> h0
    zero_kernel<<<zeroAggBlocks, T, 0, stream>>>(agg, aggElems);
    scatter_kernel<<<scatBlocks, T, 0, stream>>>(h1, src, dst, ew, norm_s, agg, E);
    wmma_gemm_kernel<D_FEAT><<<gemm64Blocks, T, 0, stream>>>(agg, W3, b3, norm_d,
                                                             nullptr, h0, N, 1);
    // Layer 4 + residual + relu: h1 = relu(gcn(h0; W4,b4) + res)
    zero_kernel<<<zeroAggBlocks, T, 0, stream>>>(agg, aggElems);
    scatter_kernel<<<scatBlocks, T, 0, stream>>>(h0, src, dst, ew, norm_s, agg, E);
    wmma_gemm_kernel<D_FEAT><<<gemm64Blocks, T, 0, stream>>>(agg, W4, b4, norm_d,
                                                             res, h1, N, 1);
    // Output projection: out = h1 @ Wo + bo
    wmma_gemm_kernel<32><<<gemmOutBlocks, T, 0, stream>>>(h1, Wo, bo, nullptr,
                                                          nullptr, (float*)d_out, N, 0);
}